// LightGCN_43353399885940
// MI455X (gfx1250) — compile-verified
//
#include <hip/hip_runtime.h>
#include <hip/hip_bf16.h>

#define EMB 64
#define WAVES_PER_BLOCK 8
#define CHUNK 32

// ---- CDNA5 async global->LDS path (gfx1250) -------------------------------
#ifndef HAS_ASYNC
#if defined(__has_builtin)
#if __has_builtin(__builtin_amdgcn_global_load_async_to_lds_b32)
#define HAS_ASYNC 1
#endif
#endif
#endif
#ifndef HAS_ASYNC
#define HAS_ASYNC 0
#endif

// Builtin expects: (global int* src, lds int* dst, imm offset, imm cpol)
typedef __attribute__((address_space(1))) int* g32_t;
typedef __attribute__((address_space(3))) int* l32_t;

// ---------------------------------------------------------------------------
// sum = emb   (vectorized copy, fully overwrites poisoned d_out)
__global__ __launch_bounds__(256) void lgcn_init_sum(const float4* __restrict__ src,
                                                     float4* __restrict__ dst, int n4) {
    int i = blockIdx.x * blockDim.x + threadIdx.x;
    if (i < n4) dst[i] = src[i];
}

// sorted-COO -> CSR row pointers via lower_bound binary search
__global__ __launch_bounds__(256) void lgcn_rowptr(const int* __restrict__ rows, int n_edges,
                                                   int* __restrict__ rowptr, int n_nodes) {
    int r = blockIdx.x * blockDim.x + threadIdx.x;
    if (r > n_nodes) return;
    int lo = 0, hi = n_edges;
    while (lo < hi) {
        int mid = (lo + hi) >> 1;
        if (rows[mid] < r) lo = mid + 1; else hi = mid;
    }
    rowptr[r] = lo;
}

// One wave32 per row: lane l accumulates embedding components [2l, 2l+1].
// Edge chunk (cols, vals) staged into LDS with CDNA5 async-to-LDS DMA,
// then consumed as wave-uniform broadcasts while lanes gather x[col] as
// coalesced 256B float2 loads (L2-resident: x fits easily in 192MB L2).
// Fuses sum accumulation: sum = (sum + y) * post_scale.
__global__ __launch_bounds__(256) void lgcn_spmm(const float* __restrict__ x,
                                                 const int* __restrict__ cols,
                                                 const float* __restrict__ vals,
                                                 const int* __restrict__ rowptr,
                                                 float* __restrict__ y,
                                                 float* __restrict__ sum,
                                                 int n_nodes, float post_scale, int write_y) {
    __shared__ int   s_col[WAVES_PER_BLOCK][CHUNK];
    __shared__ float s_val[WAVES_PER_BLOCK][CHUNK];

    const int lane = threadIdx.x & 31;
    const int w    = threadIdx.x >> 5;
    const int row  = blockIdx.x * WAVES_PER_BLOCK + w;
    if (row >= n_nodes) return;

    const int e0 = rowptr[row];
    const int e1 = rowptr[row + 1];

    float2 acc; acc.x = 0.0f; acc.y = 0.0f;

    for (int base = e0; base < e1; base += CHUNK) {
        const int n = min(CHUNK, e1 - base);
        if (lane < n) {
#if HAS_ASYNC
            __builtin_amdgcn_global_load_async_to_lds_b32(
                (g32_t)(cols + base + lane), (l32_t)(&s_col[w][lane]), 0, 0);
            __builtin_amdgcn_global_load_async_to_lds_b32(
                (g32_t)(vals + base + lane), (l32_t)(&s_val[w][lane]), 0, 0);
#else
            s_col[w][lane] = cols[base + lane];
            s_val[w][lane] = vals[base + lane];
#endif
        }
        // prefetch next edge chunk while this one is in flight
        if (base + CHUNK < e1) {
            __builtin_prefetch(cols + base + CHUNK + lane, 0, 0);
            __builtin_prefetch(vals + base + CHUNK + lane, 0, 0);
        }
#if HAS_ASYNC
        asm volatile("s_wait_asynccnt 0" ::: "memory");
#endif
        for (int j = 0; j < n; ++j) {
            const int   c = s_col[w][j];   // wave-uniform broadcast
            const float v = s_val[w][j];
            const float2 xv = ((const float2*)(x + (size_t)c * EMB))[lane];
            acc.x = fmaf(v, xv.x, acc.x);
            acc.y = fmaf(v, xv.y, acc.y);
        }
    }

    if (write_y) {
        ((float2*)(y + (size_t)row * EMB))[lane] = acc;
    }
    float2* sp = ((float2*)(sum + (size_t)row * EMB)) + lane;
    float2 sv = *sp;
    sv.x = (sv.x + acc.x) * post_scale;
    sv.y = (sv.y + acc.y) * post_scale;
    *sp = sv;
}

extern "C" void kernel_launch(void* const* d_in, const int* in_sizes, int n_in,
                              void* d_out, int out_size, void* d_ws, size_t ws_size,
                              hipStream_t stream) {
    const float* emb      = (const float*)d_in[0];
    const int*   adj_row  = (const int*)d_in[1];
    const int*   adj_col  = (const int*)d_in[2];
    const float* adj_vals = (const float*)d_in[3];

    const int n_edges = in_sizes[1];
    const int n_nodes = in_sizes[0] / EMB;   // 150000

    float* sum = (float*)d_out;              // running embs_sum lives in d_out

    // workspace layout: [rowptr (n_nodes+1 ints)] [bufA NxEMB] [bufB NxEMB]
    char* ws = (char*)d_ws;
    int* rowptr = (int*)ws;
    size_t off = (((size_t)(n_nodes + 1) * sizeof(int)) + 255) & ~(size_t)255;
    float* bufA = (float*)(ws + off);
    float* bufB = bufA + (size_t)n_nodes * EMB;

    // 1) sum = emb
    {
        int n4 = (n_nodes * EMB) / 4;
        lgcn_init_sum<<<(n4 + 255) / 256, 256, 0, stream>>>((const float4*)emb,
                                                            (float4*)sum, n4);
    }
    // 2) CSR row pointers from sorted COO rows
    lgcn_rowptr<<<(n_nodes + 1 + 255) / 256, 256, 0, stream>>>(adj_row, n_edges,
                                                               rowptr, n_nodes);

    const int grid = (n_nodes + WAVES_PER_BLOCK - 1) / WAVES_PER_BLOCK;
    // 3) layer 1: y=A*emb into bufA, sum += y
    lgcn_spmm<<<grid, 256, 0, stream>>>(emb, adj_col, adj_vals, rowptr,
                                        bufA, sum, n_nodes, 1.0f, 1);
    // 4) layer 2: y=A*bufA into bufB, sum += y
    lgcn_spmm<<<grid, 256, 0, stream>>>(bufA, adj_col, adj_vals, rowptr,
                                        bufB, sum, n_nodes, 1.0f, 1);
    // 5) layer 3: sum = (sum + A*bufB) * 0.25, skip writing y
    lgcn_spmm<<<grid, 256, 0, stream>>>(bufB, adj_col, adj_vals, rowptr,
                                        bufA, sum, n_nodes, 0.25f, 0);
}